// GraphAE_85237920956986
// MI455X (gfx1250) — compile-verified
//
#include <hip/hip_runtime.h>
#include <hip/hip_bf16.h>

#define N_NODES  131072
#define N_EDGES  262144
#define EMB      256
#define N_GRAPHS 4096
#define BN_EPS   1e-5f

typedef float v2f __attribute__((ext_vector_type(2)));
typedef float v8f __attribute__((ext_vector_type(8)));

// ---------------------------------------------------------------------------
// AtomEncoder: h[n][c] = sum_f atom_emb[x[n][f] + off[f]][c]
// ---------------------------------------------------------------------------
__global__ void atom_encode(const float* __restrict__ atom_emb,
                            const int* __restrict__ x,
                            float* __restrict__ h) {
    const int n = blockIdx.x;
    const int c = threadIdx.x;
    const int off[9] = {0, 119, 123, 135, 147, 157, 163, 168, 170};
    float s = 0.f;
#pragma unroll
    for (int f = 0; f < 9; ++f) {
        int idx = x[n * 9 + f] + off[f];
        s += atom_emb[(size_t)idx * EMB + c];
    }
    h[(size_t)n * EMB + c] = s;
}

// ---------------------------------------------------------------------------
// Degree count: atomic add 1.0 at row[e]
// ---------------------------------------------------------------------------
__global__ void degree_count(const int* __restrict__ edge_index,
                             float* __restrict__ degf) {
    int e = blockIdx.x * blockDim.x + threadIdx.x;
    if (e < N_EDGES) atomicAdd(&degf[edge_index[e]], 1.0f);
}

// deg = cnt + 1 ; dinv = deg^-0.5 ; rdeg = 1/deg
__global__ void degree_finish(float* __restrict__ degf,
                              float* __restrict__ dinv,
                              float* __restrict__ rdeg) {
    int n = blockIdx.x * blockDim.x + threadIdx.x;
    if (n < N_NODES) {
        float d = degf[n] + 1.0f;
        degf[n] = d;
        dinv[n] = rsqrtf(d);
        rdeg[n] = 1.0f / d;
    }
}

__global__ void edge_norm(const int* __restrict__ edge_index,
                          const float* __restrict__ dinv,
                          float* __restrict__ norm) {
    int e = blockIdx.x * blockDim.x + threadIdx.x;
    if (e < N_EDGES) {
        int r = edge_index[e];
        int c = edge_index[N_EDGES + e];
        norm[e] = dinv[r] * dinv[c];
    }
}

// ---------------------------------------------------------------------------
// GEMM: out = H (N x 256) @ Wm (256 x 256) + bias, fp32 via V_WMMA_F32_16X16X4_F32
// One wave handles one 16-row M-tile and a 128-column half (8 n-tiles).
// A 16x4 frag: lanes 0-15 M=0..15 {K=k0,k0+1}; lanes 16-31 {K=k0+2,k0+3}.
// B 4x16 frag: lanes = N, same K split. C/D: VGPR r = row r (lanes 0-15) /
// row r+8 (lanes 16-31), lane%16 = column.
// ---------------------------------------------------------------------------
__global__ void gemm_bias_wmma(const float* __restrict__ H,
                               const float* __restrict__ Wm,
                               const float* __restrict__ bias,
                               float* __restrict__ out) {
    const int wave  = (int)((blockIdx.x * blockDim.x + threadIdx.x) >> 5);
    const int lane  = (int)(threadIdx.x & 31);
    const int tileM = wave >> 1;     // 0..8191
    const int nhalf = wave & 1;      // which 128-column half
    const int half  = lane >> 4;     // K sub-pair selector
    const int lrow  = lane & 15;     // M for A-frag, N for B-frag

    const float* Arow = H + (size_t)(tileM * 16 + lrow) * EMB;
    const int colBase = nhalf * 128 + lrow;

    v8f c[8];
#pragma unroll
    for (int t = 0; t < 8; ++t) c[t] = (v8f)0.f;

    for (int k0 = 0; k0 < EMB; k0 += 4) {
        const int ka = k0 + 2 * half;
        v2f a = *(const v2f*)(Arow + ka);          // A[M=lrow][ka], [ka+1]
        const float* Bk = Wm + (size_t)ka * EMB + colBase;
#pragma unroll
        for (int t = 0; t < 8; ++t) {
            v2f bf;
            bf.x = Bk[t * 16];                      // B[ka][col]
            bf.y = Bk[t * 16 + EMB];                // B[ka+1][col]
            c[t] = __builtin_amdgcn_wmma_f32_16x16x4_f32(
                /*neg_a=*/false, a, /*neg_b=*/false, bf,
                /*c_mod=*/(short)0, c[t], /*reuse_a=*/false, /*reuse_b=*/false);
        }
    }

#pragma unroll
    for (int t = 0; t < 8; ++t) {
        const int col = colBase + t * 16;
        const float bv = bias[col];
#pragma unroll
        for (int r = 0; r < 8; ++r) {
            const int m = tileM * 16 + r + half * 8;
            out[(size_t)m * EMB + col] = c[t][r] + bv;
        }
    }
}

// ---------------------------------------------------------------------------
// Edge message: agg[col] += norm * relu(hl[row] + bond_emb sum)
// ---------------------------------------------------------------------------
__global__ void edge_msg(const float* __restrict__ hl,
                         const float* __restrict__ bond_emb_l,
                         const int* __restrict__ edge_index,
                         const int* __restrict__ edge_attr,
                         const float* __restrict__ norm,
                         float* __restrict__ agg) {
    const int e = blockIdx.x;
    const int c = threadIdx.x;
    const int r  = edge_index[e];
    const int cl = edge_index[N_EDGES + e];
    const float nm = norm[e];
    const int b0 = edge_attr[e * 3 + 0];       // + 0
    const int b1 = edge_attr[e * 3 + 1] + 5;
    const int b2 = edge_attr[e * 3 + 2] + 11;
    float ee = bond_emb_l[(size_t)b0 * EMB + c]
             + bond_emb_l[(size_t)b1 * EMB + c]
             + bond_emb_l[(size_t)b2 * EMB + c];
    float m = hl[(size_t)r * EMB + c] + ee;
    m = fmaxf(m, 0.f) * nm;
    atomicAdd(&agg[(size_t)cl * EMB + c], m);
}

// ---------------------------------------------------------------------------
// Node update: h = agg + relu(hl + root)/deg, then eval BatchNorm (+ReLU)
// ---------------------------------------------------------------------------
__global__ void node_update(const float* __restrict__ agg,
                            const float* __restrict__ hl,
                            const float* __restrict__ root_l,
                            const float* __restrict__ gamma,
                            const float* __restrict__ beta,
                            const float* __restrict__ mean,
                            const float* __restrict__ var,
                            const float* __restrict__ rdeg,
                            float* __restrict__ h,
                            int relu_out) {
    const int n = blockIdx.x;
    const int c = threadIdx.x;
    float v = agg[(size_t)n * EMB + c]
            + fmaxf(hl[(size_t)n * EMB + c] + root_l[c], 0.f) * rdeg[n];
    v = (v - mean[c]) * (gamma[c] * rsqrtf(var[c] + BN_EPS)) + beta[c];
    if (relu_out) v = fmaxf(v, 0.f);
    h[(size_t)n * EMB + c] = v;
}

// ---------------------------------------------------------------------------
// global_add_pool: out[batch[n]] += h[n]
// ---------------------------------------------------------------------------
__global__ void pool_add(const float* __restrict__ h,
                         const int* __restrict__ batch,
                         float* __restrict__ out) {
    const int n = blockIdx.x;
    const int c = threadIdx.x;
    const int g = batch[n];
    atomicAdd(&out[(size_t)g * EMB + c], h[(size_t)n * EMB + c]);
}

// ---------------------------------------------------------------------------
extern "C" void kernel_launch(void* const* d_in, const int* in_sizes, int n_in,
                              void* d_out, int out_size, void* d_ws, size_t ws_size,
                              hipStream_t stream) {
    const float* atom_emb = (const float*)d_in[0];   // [172,256]
    const float* bond_emb = (const float*)d_in[1];   // [3,13,256]
    const float* W        = (const float*)d_in[2];   // [3,256,256]
    const float* b        = (const float*)d_in[3];   // [3,256]
    const float* root     = (const float*)d_in[4];   // [3,256]
    const float* bn_gamma = (const float*)d_in[5];   // [3,256]
    const float* bn_beta  = (const float*)d_in[6];
    const float* bn_mean  = (const float*)d_in[7];
    const float* bn_var   = (const float*)d_in[8];
    const int*   x        = (const int*)d_in[9];     // [131072,9]
    const int*   edge_idx = (const int*)d_in[10];    // [2,262144]
    const int*   edge_atr = (const int*)d_in[11];    // [262144,3]
    const int*   batch    = (const int*)d_in[12];    // [131072]
    float* out = (float*)d_out;                      // [4096,256]

    // Workspace layout (floats)
    const size_t NE = (size_t)N_NODES * EMB;         // 33,554,432
    float* h    = (float*)d_ws;                      // 128 MB
    float* hl   = h    + NE;                         // 128 MB
    float* agg  = hl   + NE;                         // 128 MB
    float* degf = agg  + NE;                         // 0.5 MB
    float* dinv = degf + N_NODES;
    float* rdeg = dinv + N_NODES;
    float* nrm  = rdeg + N_NODES;                    // 1 MB

    // ---- encoder + degree/norm precompute ----
    atom_encode<<<N_NODES, EMB, 0, stream>>>(atom_emb, x, h);

    hipMemsetAsync(degf, 0, (size_t)N_NODES * sizeof(float), stream);
    degree_count<<<(N_EDGES + 255) / 256, 256, 0, stream>>>(edge_idx, degf);
    degree_finish<<<(N_NODES + 255) / 256, 256, 0, stream>>>(degf, dinv, rdeg);
    edge_norm<<<(N_EDGES + 255) / 256, 256, 0, stream>>>(edge_idx, dinv, nrm);

    // ---- layers ----
    const int gemmBlocks = (8192 /*M tiles*/ * 2 /*col halves*/ * 32) / 256; // 2048
    for (int l = 0; l < 3; ++l) {
        const float* W_l    = W        + (size_t)l * EMB * EMB;
        const float* b_l    = b        + (size_t)l * EMB;
        const float* root_l = root     + (size_t)l * EMB;
        const float* g_l    = bn_gamma + (size_t)l * EMB;
        const float* be_l   = bn_beta  + (size_t)l * EMB;
        const float* m_l    = bn_mean  + (size_t)l * EMB;
        const float* v_l    = bn_var   + (size_t)l * EMB;
        const float* bond_l = bond_emb + (size_t)l * 13 * EMB;

        gemm_bias_wmma<<<gemmBlocks, 256, 0, stream>>>(h, W_l, b_l, hl);

        hipMemsetAsync(agg, 0, NE * sizeof(float), stream);
        edge_msg<<<N_EDGES, EMB, 0, stream>>>(hl, bond_l, edge_idx, edge_atr, nrm, agg);

        node_update<<<N_NODES, EMB, 0, stream>>>(agg, hl, root_l, g_l, be_l, m_l, v_l,
                                                 rdeg, h, (l != 2) ? 1 : 0);
    }

    // ---- pooling ----
    hipMemsetAsync(out, 0, (size_t)N_GRAPHS * EMB * sizeof(float), stream);
    pool_add<<<N_NODES, EMB, 0, stream>>>(h, batch, out);
}